// PointnetSampleGroup_73787538145797
// MI455X (gfx1250) — compile-verified
//
#include <hip/hip_runtime.h>
#include <hip/hip_bf16.h>

// Problem constants (from reference)
#define Bq   16
#define Nk   4096
#define Cf   256
#define Mq   1024
#define Sn   64
#define RAD  0.2f
#define RAD2 (RAD * RAD)

typedef float v2f __attribute__((ext_vector_type(2)));
typedef float v4f __attribute__((ext_vector_type(4)));
typedef float v8f __attribute__((ext_vector_type(8)));

// ---------------------------------------------------------------------------
// Kernel 1: ball query via V_WMMA_F32_16X16X4_F32, keys staged in LDS.
// Block = 128 threads (4 waves) = 64 queries, all from one batch.
// The batch's 4096 keys (48 KB) are cooperatively staged into LDS once,
// then each wave's 256-tile scan reads them with ds_load_b96 (no global
// round-trip stall per tile). One wave handles 16 query points:
//   A (16x4) = query coords (x,y in lanes 0-15 / z,0 in lanes 16-31)
//   B (4x16) = key   coords (same striping)
//   D (16x16 f32, 8 VGPRs) = q.k dot products
// d2 = |q|^2 + |k|^2 - 2*dot ; ballot per VGPR row -> 16-bit hit mask whose
// bit order == key index order; lanes 0-15 compact (ctz) into the index list.
// ---------------------------------------------------------------------------
__global__ __launch_bounds__(128) void ball_query_wmma_kernel(
    const float* __restrict__ key_xyz,     // (B, N, 3)
    const float* __restrict__ query_xyz,   // (B, M, 6)
    int* __restrict__ idx_out)             // (B*M, 64) in workspace
{
  __shared__ float skeys[Nk * 3];          // 48 KB of 320 KB/WGP

  const int t    = threadIdx.x;
  const int lane = t & 31;
  const int col  = lane & 15;
  const bool hi  = (lane >= 16);

  const int qrow0_blk = blockIdx.x * 64;   // 64 queries per block
  const int b         = qrow0_blk / Mq;    // 64 | M, so block stays in batch
  const int qrow0     = qrow0_blk + (t >> 5) * 16;

  // --- Cooperative stage: batch keys -> LDS (3072 x b128) ------------------
  {
    const v4f* __restrict__ src = (const v4f*)(key_xyz + (size_t)b * Nk * 3);
    v4f* dst = (v4f*)skeys;
#pragma unroll
    for (int i = 0; i < (Nk * 3 / 4) / 128; ++i)   // 24 iterations
      dst[t + i * 128] = src[t + i * 128];
  }

  // --- Load A matrix (queries), precompute |q|^2 per row -------------------
  const float* qp = query_xyz + (size_t)(qrow0 + col) * 6;
  const float qx = qp[0], qy = qp[1], qz = qp[2];
  const float qq = qx * qx + qy * qy + qz * qz;

  v2f amat;
  amat.x = hi ? qz : qx;
  amat.y = hi ? 0.0f : qy;

  // qq_r[r] = |q|^2 of the row this lane sees in D-vgpr r (row = r + 8*hi)
  float qq_r[8];
#pragma unroll
  for (int r = 0; r < 8; ++r)
    qq_r[r] = __shfl(qq, r + (hi ? 8 : 0), 32);

  __syncthreads();                         // keys resident in LDS

  // Lane L < 16 owns query row qrow0+L (never dereferenced for lanes >= 16).
  int* idxrow = idx_out + (size_t)(qrow0 + (lane & 15)) * Sn;

  int cnt   = (lane < 16) ? 0 : Sn;  // lanes >=16 report "full" for the ballot
  int first = 0;

  for (int tile = 0; tile < Nk / 16; ++tile) {
    // Uniform early exit once every row has 64 neighbors.
    if (__builtin_amdgcn_ballot_w32(cnt < Sn) == 0u) break;

    const int nb = tile * 16;
    const float* kp = skeys + (size_t)(nb + col) * 3;   // ds_load_b96
    const float kx = kp[0], ky = kp[1], kz = kp[2];
    const float kk = kx * kx + ky * ky + kz * kz;       // |k|^2 of column

    v2f bmat;
    bmat.x = hi ? kz : kx;
    bmat.y = hi ? 0.0f : ky;

    v8f cz = {};
    v8f d = __builtin_amdgcn_wmma_f32_16x16x4_f32(
        /*neg_a=*/false, amat, /*neg_b=*/false, bmat,
        /*c_mod=*/(short)0, cz, /*reuse_a=*/false, /*reuse_b=*/false);

    unsigned msk[8];
#pragma unroll
    for (int r = 0; r < 8; ++r) {
      const float d2 = qq_r[r] + kk - 2.0f * d[r];
      msk[r] = __builtin_amdgcn_ballot_w32(d2 < RAD2); // wave-uniform result
    }

    // Row L: 32-bit word msk[L&7]; halves: L<8 -> [15:0], L>=8 -> [31:16].
    unsigned m32 = 0u;
#pragma unroll
    for (int r = 0; r < 8; ++r)
      if ((lane & 7) == r) m32 = msk[r];
    unsigned rowmask =
        (lane < 16) ? ((lane & 8) ? (m32 >> 16) : (m32 & 0xffffu)) : 0u;

    while (rowmask != 0u && cnt < Sn) {
      const int bit  = __builtin_ctz(rowmask);
      rowmask &= rowmask - 1u;
      const int nidx = nb + bit;
      if (cnt == 0) first = nidx;
      idxrow[cnt] = nidx;
      ++cnt;
    }
  }

  // Pad remaining slots with the first hit (0 if no hit), matching reference.
  if (lane < 16) {
    for (int k = cnt; k < Sn; ++k) idxrow[k] = first;
  }
}

// ---------------------------------------------------------------------------
// Kernel 2: gather. One block per (b, m). Indices + query xyz staged in LDS.
// grouped_xyz: threads 0..191. Features: thread t owns channel ch = t>>4 of
// each 16-channel group and 4 consecutive samples -> 128-bit coalesced
// NON-TEMPORAL stores (1.07 GB streamed out, never re-read) so the 64 MB
// key_features working set stays L2-resident for the gather reads.
// ---------------------------------------------------------------------------
__global__ __launch_bounds__(256) void gather_group_kernel(
    const float* __restrict__ key_xyz,     // (B, N, 3)
    const float* __restrict__ key_feat,    // (B, C, N)
    const float* __restrict__ query_xyz,   // (B, M, 6)
    const int* __restrict__ idx_in,        // (B*M, 64)
    float* __restrict__ out_xyz,           // (B, 3, M, 64)
    float* __restrict__ out_feat)          // (B, C, M, 64)
{
  const int bm = blockIdx.x;       // b*M + m
  const int b  = bm >> 10;         // / Mq
  const int m  = bm & (Mq - 1);
  const int t  = threadIdx.x;

  __shared__ int   sidx[Sn];
  __shared__ float sq[3];

  if (t < Sn) sidx[t] = idx_in[(size_t)bm * Sn + t];
  if (t < 3)  sq[t]   = query_xyz[(size_t)bm * 6 + t];
  __syncthreads();

  // grouped_xyz = (key - query) / RADIUS, layout (B,3,M,S)
  if (t < 3 * Sn) {
    const int d = t >> 6, s = t & 63;
    const int n = sidx[s];
    const float v = (key_xyz[((size_t)b * Nk + n) * 3 + d] - sq[d]) * (1.0f / RAD);
    __builtin_nontemporal_store(
        v, out_xyz + (((size_t)b * 3 + d) * Mq + m) * Sn + s);
  }

  // grouped_features, layout (B,C,M,S)
  const int ch = t >> 4;           // 0..15 : channel within 16-channel group
  const int s0 = (t & 15) * 4;     // 4 consecutive samples
  const int i0 = sidx[s0], i1 = sidx[s0 + 1], i2 = sidx[s0 + 2], i3 = sidx[s0 + 3];
  const float* fb = key_feat + (size_t)b * Cf * Nk;

#pragma unroll 4
  for (int cb = 0; cb < Cf / 16; ++cb) {
    const int c = cb * 16 + ch;
    const float* row = fb + (size_t)c * Nk;              // 16 KB row, L2 hot
    v4f v = {row[i0], row[i1], row[i2], row[i3]};
    __builtin_nontemporal_store(                         // b128 NT store
        v, (v4f*)(out_feat + (((size_t)b * Cf + c) * Mq + m) * Sn + s0));
  }
}

// ---------------------------------------------------------------------------
extern "C" void kernel_launch(void* const* d_in, const int* in_sizes, int n_in,
                              void* d_out, int out_size, void* d_ws, size_t ws_size,
                              hipStream_t stream) {
  const float* key_xyz   = (const float*)d_in[0]; // (16,4096,3)
  const float* key_feat  = (const float*)d_in[1]; // (16,256,4096)
  const float* query_xyz = (const float*)d_in[2]; // (16,1024,6)

  float* out_xyz  = (float*)d_out;                            // (16,3,1024,64)
  float* out_feat = out_xyz + (size_t)Bq * 3 * Mq * Sn;       // (16,256,1024,64)

  int* idx_ws = (int*)d_ws;  // needs B*M*64*4 = 4 MB of workspace

  // 256 blocks x 4 waves, 16 queries per wave.
  ball_query_wmma_kernel<<<(Bq * Mq) / 64, 128, 0, stream>>>(
      key_xyz, query_xyz, idx_ws);

  // One block per (b, m) = 16384 blocks.
  gather_group_kernel<<<Bq * Mq, 256, 0, stream>>>(
      key_xyz, key_feat, query_xyz, idx_ws, out_xyz, out_feat);
}